// GraphConvolution2_w_57389353009529
// MI455X (gfx1250) — compile-verified
//
#include <hip/hip_runtime.h>
#include <hip/hip_bf16.h>

typedef __attribute__((ext_vector_type(2))) float v2f;
typedef __attribute__((ext_vector_type(4))) float v4f;
typedef __attribute__((ext_vector_type(8))) float v8f;

#define NDIM 4096
#define FDIM 256
#define MP1  11
#define HK_THRESH 1e-5f

// -------------------------------------------------------------------------
// Kernel 1: one streaming pass over L_n (11 x 4096 x 4096 f32, 738 MB).
// Each thread owns 4 consecutive q for a fixed row p (16B vector loads).
// Computes lag[p,q] (row-coeff weighted sum) and lag_grad[p,q] (col-coeff
// weighted sum), applies the threshold mask, writes lag -> workspace (RT,
// so it stays L2-resident for the WMMA GEMM) and lag_grad -> output (NT,
// write-once).  L_n loads are non-temporal (read-once stream, 738 MB >>
// 192 MB L2 -- don't let it evict lag).
//   coeff[n]      = (1/(t+1)) * (t/(t+1))^n
//   coeff_grad[n] = coeff[n] * (n - t) / (t*(t+1))
// -------------------------------------------------------------------------
__global__ void __launch_bounds__(256)
hk_fuse_kernel(const float* __restrict__ Ln, const float* __restrict__ t,
               float* __restrict__ lag, float* __restrict__ lag_grad)
{
    const int p  = blockIdx.y;
    const int q0 = blockIdx.x * 1024 + threadIdx.x * 4;

    // row coefficients (uniform across the block -> scalarizable)
    const float tp = t[p];
    const float rp = tp / (tp + 1.0f);
    float coeffP[MP1];
    {
        float c = 1.0f / (tp + 1.0f);
#pragma unroll
        for (int n = 0; n < MP1; ++n) { coeffP[n] = c; c *= rp; }
    }

    // column-gradient coefficients for this thread's 4 q values
    const v4f tq = *(const v4f*)(t + q0);
    float cg[MP1][4];
#pragma unroll
    for (int j = 0; j < 4; ++j) {
        const float tj  = tq[j];
        const float r   = tj / (tj + 1.0f);
        const float inv = 1.0f / (tj * (tj + 1.0f));
        float c = 1.0f / (tj + 1.0f);
#pragma unroll
        for (int n = 0; n < MP1; ++n) {
            cg[n][j] = c * ((float)n - tj) * inv;
            c *= r;
        }
    }

    v4f aL = {0.f, 0.f, 0.f, 0.f};
    v4f aG = {0.f, 0.f, 0.f, 0.f};
    const size_t base = (size_t)p * NDIM + (size_t)q0;
#pragma unroll
    for (int n = 0; n < MP1; ++n) {
        const v4f Lv = __builtin_nontemporal_load(
            (const v4f*)(Ln + (size_t)n * (size_t)NDIM * NDIM + base));
        aL += coeffP[n] * Lv;
        aG[0] += cg[n][0] * Lv[0];
        aG[1] += cg[n][1] * Lv[1];
        aG[2] += cg[n][2] * Lv[2];
        aG[3] += cg[n][3] * Lv[3];
    }

    v4f oL, oG;
#pragma unroll
    for (int j = 0; j < 4; ++j) {
        const bool m = (aL[j] >= HK_THRESH);
        oL[j] = m ? aL[j] : 0.f;
        oG[j] = m ? aG[j] : 0.f;
    }

    *(v4f*)(lag + base) = oL;                                    // RT: keep in L2
    __builtin_nontemporal_store(oG, (v4f*)(lag_grad + base));    // NT: write-once
}

// -------------------------------------------------------------------------
// Kernel 2: out(4096x256) = lag(4096x4096) @ x(4096x256), fp32 WMMA.
// One wave computes a 64(M) x 16(N) strip: 4 accumulators sharing each
// B fragment.  K advances in steps of 4 via V_WMMA_F32_16X16X4_F32.
//   A 16x4:  lane<16 -> (A[m][k],A[m][k+1]); lane>=16 -> (A[m][k+2],A[m][k+3])
//   B 4x16:  lane<16 -> (B[k][n],B[k+1][n]); lane>=16 -> (B[k+2][n],B[k+3][n])
//   C/D:     VGPR r -> rows r (lanes 0-15) and r+8 (lanes 16-31)
// lag and x both live in L2 during this kernel (64 MB + 4 MB << 192 MB).
// -------------------------------------------------------------------------
__global__ void __launch_bounds__(256)
hk_gemm_f32_kernel(const float* __restrict__ lag, const float* __restrict__ x,
                   float* __restrict__ out)
{
    const int wave  = (blockIdx.x * blockDim.x + threadIdx.x) >> 5;
    const int lane  = threadIdx.x & 31;
    const int mTile = wave >> 4;        // 64 strips of 64 rows
    const int nTile = wave & 15;        // 16 tiles of 16 cols
    const int m0 = mTile * 64;
    const int n0 = nTile * 16;
    const int lm = lane & 15;
    const int lh = lane >> 4;           // 0: K,K+1   1: K+2,K+3

    v8f acc[4] = {};
    const float* bp = x + n0 + lm;

    for (int k = 0; k < NDIM; k += 4) {
        v2f B;
        B.x = bp[(size_t)(k + 2 * lh) * FDIM];
        B.y = bp[(size_t)(k + 2 * lh + 1) * FDIM];
#pragma unroll
        for (int i = 0; i < 4; ++i) {
            const float* ap = lag + (size_t)(m0 + i * 16 + lm) * NDIM + k + 2 * lh;
            v2f A = *(const v2f*)ap;
            acc[i] = __builtin_amdgcn_wmma_f32_16x16x4_f32(
                false, A, false, B, (short)0, acc[i], false, false);
        }
    }

#pragma unroll
    for (int i = 0; i < 4; ++i) {
#pragma unroll
        for (int r = 0; r < 8; ++r) {
            out[(size_t)(m0 + i * 16 + 8 * lh + r) * FDIM + n0 + lm] = acc[i][r];
        }
    }
}

extern "C" void kernel_launch(void* const* d_in, const int* in_sizes, int n_in,
                              void* d_out, int out_size, void* d_ws, size_t ws_size,
                              hipStream_t stream) {
    (void)in_sizes; (void)n_in; (void)out_size; (void)ws_size;
    const float* x  = (const float*)d_in[0];   // (4096, 256)
    const float* Ln = (const float*)d_in[1];   // (11, 4096, 4096)
    const float* t  = (const float*)d_in[2];   // (4096,)
    // d_in[3] = eigmax, unused by the laguerre branch

    float* out      = (float*)d_out;                       // (4096, 256)
    float* lag_grad = out + (size_t)NDIM * FDIM;           // (4096, 4096)
    float* lag      = (float*)d_ws;                        // 64 MiB scratch

    // Pass 1: stream L_n once (NT), produce lag (L2-resident) + lag_grad (NT)
    hk_fuse_kernel<<<dim3(4, NDIM), 256, 0, stream>>>(Ln, t, lag, lag_grad);

    // Pass 2: out = lag @ x with fp32 WMMA.
    // waves = (4096/64) * (256/16) = 1024 -> 128 blocks of 8 waves
    hk_gemm_f32_kernel<<<128, 256, 0, stream>>>(lag, x, out);
}